// Masking_65068754534997
// MI455X (gfx1250) — compile-verified
//
#include <hip/hip_runtime.h>
#include <stdint.h>

#define IMG_H 2048
#define IMG_W 4096
#define NBLK  2048            // 32 * 64 blocks of 64x64
#define NPIX  4096            // pixels per block
#define THREADS 256

typedef __attribute__((ext_vector_type(2))) float v2f;
typedef __attribute__((ext_vector_type(8))) float v8f;

// ---------------- threefry2x32 (exact JAX PRNG) ----------------
__host__ __device__ static inline void tf2x32(unsigned k0, unsigned k1,
                                              unsigned c0, unsigned c1,
                                              unsigned &o0, unsigned &o1) {
  unsigned ks2 = k0 ^ k1 ^ 0x1BD11BDAu;
  unsigned x0 = c0 + k0;
  unsigned x1 = c1 + k1;
#define TF_R(x, r) (((x) << (r)) | ((x) >> (32 - (r))))
#define TF_RND(r) { x0 += x1; x1 = TF_R(x1, r); x1 ^= x0; }
  TF_RND(13) TF_RND(15) TF_RND(26) TF_RND(6)
  x0 += k1;  x1 += ks2 + 1u;
  TF_RND(17) TF_RND(29) TF_RND(16) TF_RND(24)
  x0 += ks2; x1 += k0 + 2u;
  TF_RND(13) TF_RND(15) TF_RND(26) TF_RND(6)
  x0 += k0;  x1 += k1 + 3u;
  TF_RND(17) TF_RND(29) TF_RND(16) TF_RND(24)
  x0 += k1;  x1 += ks2 + 4u;
  TF_RND(13) TF_RND(15) TF_RND(26) TF_RND(6)
  x0 += ks2; x1 += k0 + 5u;
  o0 = x0; o1 = x1;
#undef TF_RND
#undef TF_R
}

__device__ static inline float u01_from_bits(unsigned bits) {
  return __uint_as_float((bits >> 9) | 0x3f800000u) - 1.0f;
}

// ---------------- kernel 0: global max of depth (for max()>1.0 branch) ----
__global__ void depth_max_kernel(const float* __restrict__ depth,
                                 unsigned n4, unsigned* __restrict__ outbits) {
  float m = 0.0f;
  const float4* d4 = (const float4*)depth;
  for (unsigned i = blockIdx.x * blockDim.x + threadIdx.x; i < n4;
       i += gridDim.x * blockDim.x) {
    float4 v = d4[i];
    m = fmaxf(m, fmaxf(fmaxf(v.x, v.y), fmaxf(v.z, v.w)));
  }
  for (int off = 16; off > 0; off >>= 1)
    m = fmaxf(m, __shfl_xor(m, off, 32));
  if ((threadIdx.x & 31) == 0)
    atomicMax(outbits, __float_as_uint(m));   // floats are non-negative
}

// ---------------- kernel 1: per-block mask + apply ----------------
// LDS layout (dynamic, 54336 bytes):
//   [0      , 32768) : staged img channels 0..1 (async b128 copies)
//   [32768  , 49152) : 4096 x u32 noise keys (23-bit)
//   [49152  , 50176) : 256 x u32 histogram / inclusive scan
//   [50176  , 54272) : 4096 x u8 zero flags
//   [54272  , 54304) : 8 x f32 wave partial sums
//   [54304  , 54320) : misc ints {nz, B, cntless}
__global__ void __launch_bounds__(THREADS)
mask_apply_kernel(const float* __restrict__ img,
                  const float* __restrict__ depth,
                  float* __restrict__ out,
                  unsigned k1a, unsigned k1b, unsigned k2a, unsigned k2b,
                  const unsigned* __restrict__ maxbits) {
  extern __shared__ unsigned char smem[];
  float*         s_img  = (float*)(smem);
  unsigned*      s_key  = (unsigned*)(smem + 32768);
  unsigned*      s_hist = (unsigned*)(smem + 49152);
  unsigned char* s_zero = (unsigned char*)(smem + 50176);
  float*         s_wsum = (float*)(smem + 54272);
  int*           s_misc = (int*)(smem + 54304);

  const int tid = threadIdx.x;
  const int bf  = blockIdx.x;          // 0..2047, = by*64+bx
  const int by  = bf >> 6;
  const int bx  = bf & 63;
  const size_t baseY = (size_t)by * 64;
  const size_t baseX = (size_t)bx * 64;

  // ---- 1) async-stage img channels 0..1 into LDS (overlaps everything) ----
  for (int q = tid; q < 2048; q += THREADS) {          // 8 x b128 per thread
    int c = q >> 10, rem = q & 1023, py = rem >> 4, qx = rem & 15;
    const float* gp = img + ((size_t)(c * IMG_H) + baseY + py) * IMG_W
                          + baseX + (size_t)qx * 4;
    unsigned long long ga = (unsigned long long)(uintptr_t)gp;
    unsigned lds = (unsigned)(uintptr_t)(s_img + (size_t)q * 4);
    asm volatile("global_load_async_to_lds_b128 %0, %1, off"
                 :: "v"(lds), "v"(ga) : "memory");
  }
  // prefetch channel 2 (consumed at the end from L2/L0)
  for (int q = 2048 + tid; q < 3072; q += THREADS) {
    int rem = q & 1023, py = rem >> 4, qx = rem & 15;
    __builtin_prefetch(img + ((size_t)(2 * IMG_H) + baseY + py) * IMG_W
                           + baseX + (size_t)qx * 4, 0, 3);
  }

  // ---- 2) block depth sum over 3 channels, on the XDL pipe ----
  // D = A x ones + C  =>  every D entry is a row-sum of A; sum(D) = 16*sum(A).
  // Each v_wmma_f32_16x16x4_f32 consumes 2 floats/lane (A = 16x4 striped
  // across the wave); layout inside A is irrelevant for a full-tile sum.
  {
    v8f acc = {};
    const v2f ones = {1.0f, 1.0f};
    for (int q = tid; q < 3072; q += THREADS) {
      int c = q >> 10, rem = q & 1023, py = rem >> 4, qx = rem & 15;
      float4 v = *(const float4*)(depth + ((size_t)(c * IMG_H) + baseY + py) * IMG_W
                                        + baseX + (size_t)qx * 4);
      v2f a0 = {v.x, v.y};
      v2f a1 = {v.z, v.w};
      acc = __builtin_amdgcn_wmma_f32_16x16x4_f32(false, a0, false, ones,
                                                  (short)0, acc, false, false);
      acc = __builtin_amdgcn_wmma_f32_16x16x4_f32(false, a1, false, ones,
                                                  (short)0, acc, false, false);
    }
    float sum = ((acc[0] + acc[1]) + (acc[2] + acc[3]))
              + ((acc[4] + acc[5]) + (acc[6] + acc[7]));
    for (int off = 16; off > 0; off >>= 1) sum += __shfl_xor(sum, off, 32);
    if ((tid & 31) == 0) s_wsum[tid >> 5] = sum * (1.0f / 16.0f);  // exact /16
  }
  __syncthreads();
  if (tid == 0) {
    float t = 0.0f;
    for (int i = 0; i < 8; ++i) t += s_wsum[i];
    float mx = __uint_as_float(*maxbits);
    float scale = (mx > 1.0f) ? (1.0f / 255.0f) : 1.0f;
    // num_zeros = rint(avg_depth*4096) = rint(sum*scale/3), banker's rounding
    int nz = (int)rintf(t * scale * (1.0f / 3.0f));
    nz = nz < 0 ? 0 : (nz > NPIX ? NPIX : nz);
    s_misc[0] = nz; s_misc[1] = -1; s_misc[2] = 0;
  }
  __syncthreads();
  const int nz = s_misc[0];

  // ---- 3) keep flag: uniform(k1)[bf] > 0.5  (threefry counter layout) ----
  unsigned kb;
  { unsigned o0, o1;
    if (bf < 1024) { tf2x32(k1a, k1b, (unsigned)bf, (unsigned)bf + 1024u, o0, o1); kb = o0; }
    else           { tf2x32(k1a, k1b, (unsigned)bf - 1024u, (unsigned)bf, o0, o1); kb = o1; }
  }
  const bool keepB = (u01_from_bits(kb) > 0.5f);
  const bool sel = (!keepB) && (nz > 0);   // uniform across the block

  // ---- 4) exact stable-rank selection of nz smallest noise values ----
  if (sel) {
    s_hist[tid] = 0u;
    __syncthreads();
    unsigned mykey[16];
    const unsigned half = 4194304u;               // (32*64*4096)/2
    const unsigned gbase = (unsigned)bf * 4096u;
    for (int i = 0; i < 16; ++i) {
      unsigned e = (unsigned)tid * 16u + (unsigned)i;
      unsigned g = gbase + e;
      unsigned o0, o1, b;
      if (g < half) { tf2x32(k2a, k2b, g, g + half, o0, o1); b = o0; }
      else          { tf2x32(k2a, k2b, g - half, g, o0, o1); b = o1; }
      unsigned key = b >> 9;                      // 23-bit, monotone with float
      mykey[i] = key;
      s_key[e] = key;
      atomicAdd(&s_hist[key >> 15], 1u);
    }
    __syncthreads();
    // inclusive Hillis-Steele scan over 256 bins
    for (int off = 1; off < 256; off <<= 1) {
      unsigned t = 0u;
      if (tid >= off) t = s_hist[tid - off];
      __syncthreads();
      if (tid >= off) s_hist[tid] += t;
      __syncthreads();
    }
    unsigned cum  = s_hist[tid];
    unsigned cump = (tid > 0) ? s_hist[tid - 1] : 0u;
    if ((int)cump < nz && (int)cum >= nz) { s_misc[1] = tid; s_misc[2] = (int)cump; }
    __syncthreads();
    const int B = s_misc[1];
    for (int i = 0; i < 16; ++i) {
      int e = tid * 16 + i;
      unsigned key = mykey[i];
      int bin = (int)(key >> 15);
      unsigned char z;
      if (bin < B)      z = 1;                    // rank < cntless <= nz
      else if (bin > B) z = 0;                    // rank >= cum[B] >= nz
      else {
        int rank = 0;                             // exact stable global rank
        for (int j = 0; j < NPIX; ++j) {
          unsigned kj = s_key[j];
          rank += (kj < key) || (kj == key && j < e);
        }
        z = (rank < nz) ? 1 : 0;
      }
      s_zero[e] = z;
    }
    __syncthreads();
  }

  // ---- 5) wait for async staging, then select & store ----
  asm volatile("s_wait_asynccnt 0" ::: "memory");
  __syncthreads();
  for (int q = tid; q < 3072; q += THREADS) {
    int c = q >> 10, rem = q & 1023, py = rem >> 4, qx = rem & 15;
    size_t goff = ((size_t)(c * IMG_H) + baseY + py) * IMG_W + baseX + (size_t)qx * 4;
    float4 v = (c < 2) ? *(const float4*)(s_img + (size_t)q * 4)
                       : *(const float4*)(img + goff);
    if (sel) {
      int e0 = py * 64 + qx * 4;
      if (s_zero[e0 + 0]) v.x = 128.0f;
      if (s_zero[e0 + 1]) v.y = 128.0f;
      if (s_zero[e0 + 2]) v.z = 128.0f;
      if (s_zero[e0 + 3]) v.w = 128.0f;
    }
    *(float4*)(out + goff) = v;
  }
}

// ---------------- host ----------------
extern "C" void kernel_launch(void* const* d_in, const int* in_sizes, int n_in,
                              void* d_out, int out_size, void* d_ws, size_t ws_size,
                              hipStream_t stream) {
  (void)in_sizes; (void)n_in; (void)out_size; (void)ws_size;
  const float* img   = (const float*)d_in[0];
  const float* depth = (const float*)d_in[1];
  float* out = (float*)d_out;
  unsigned* maxws = (unsigned*)d_ws;

  // key(42) = (0,42); split -> k1 = (o0(0,2), o0(1,3)), k2 = (o1(0,2), o1(1,3))
  unsigned a0, b0, a1, b1;
  tf2x32(0u, 42u, 0u, 2u, a0, b0);
  tf2x32(0u, 42u, 1u, 3u, a1, b1);

  hipMemsetAsync(d_ws, 0, sizeof(unsigned), stream);
  depth_max_kernel<<<1024, THREADS, 0, stream>>>(depth, (3u * IMG_H * IMG_W) / 4u, maxws);
  mask_apply_kernel<<<NBLK, THREADS, 54336, stream>>>(img, depth, out,
                                                      a0, a1, b0, b1, maxws);
}